// MonoRecModel_15865609191856
// MI455X (gfx1250) — compile-verified
//
#include <hip/hip_runtime.h>
#include <math.h>

// Problem constants (match reference)
#define B_ 2
#define F_ 2
#define C_ 3
#define H_ 256
#define W_ 512
#define D_ 32
#define TIL 16
#define HALO 20   // tile + 2 halo each side (SSIM avg3 + boxsum3)
#define DWN 18    // tile + 1 halo each side (diff positions)

typedef float v2f __attribute__((ext_vector_type(2)));
typedef float v8f __attribute__((ext_vector_type(8)));

// ---------------------------------------------------------------- small linalg
__device__ __forceinline__ void inv4x4(const float* m, float* o) {
    float inv[16];
    inv[0]  =  m[5]*m[10]*m[15]-m[5]*m[11]*m[14]-m[9]*m[6]*m[15]+m[9]*m[7]*m[14]+m[13]*m[6]*m[11]-m[13]*m[7]*m[10];
    inv[4]  = -m[4]*m[10]*m[15]+m[4]*m[11]*m[14]+m[8]*m[6]*m[15]-m[8]*m[7]*m[14]-m[12]*m[6]*m[11]+m[12]*m[7]*m[10];
    inv[8]  =  m[4]*m[9]*m[15]-m[4]*m[11]*m[13]-m[8]*m[5]*m[15]+m[8]*m[7]*m[13]+m[12]*m[5]*m[11]-m[12]*m[7]*m[9];
    inv[12] = -m[4]*m[9]*m[14]+m[4]*m[10]*m[13]+m[8]*m[5]*m[14]-m[8]*m[6]*m[13]-m[12]*m[5]*m[10]+m[12]*m[6]*m[9];
    inv[1]  = -m[1]*m[10]*m[15]+m[1]*m[11]*m[14]+m[9]*m[2]*m[15]-m[9]*m[3]*m[14]-m[13]*m[2]*m[11]+m[13]*m[3]*m[10];
    inv[5]  =  m[0]*m[10]*m[15]-m[0]*m[11]*m[14]-m[8]*m[2]*m[15]+m[8]*m[3]*m[14]+m[12]*m[2]*m[11]-m[12]*m[3]*m[10];
    inv[9]  = -m[0]*m[9]*m[15]+m[0]*m[11]*m[13]+m[8]*m[1]*m[15]-m[8]*m[3]*m[13]-m[12]*m[1]*m[11]+m[12]*m[3]*m[9];
    inv[13] =  m[0]*m[9]*m[14]-m[0]*m[10]*m[13]-m[8]*m[1]*m[14]+m[8]*m[2]*m[13]+m[12]*m[1]*m[10]-m[12]*m[2]*m[9];
    inv[2]  =  m[1]*m[6]*m[15]-m[1]*m[7]*m[14]-m[5]*m[2]*m[15]+m[5]*m[3]*m[14]+m[13]*m[2]*m[7]-m[13]*m[3]*m[6];
    inv[6]  = -m[0]*m[6]*m[15]+m[0]*m[7]*m[14]+m[4]*m[2]*m[15]-m[4]*m[3]*m[14]-m[12]*m[2]*m[7]+m[12]*m[3]*m[6];
    inv[10] =  m[0]*m[5]*m[15]-m[0]*m[7]*m[13]-m[4]*m[1]*m[15]+m[4]*m[3]*m[13]+m[12]*m[1]*m[7]-m[12]*m[3]*m[5];
    inv[14] = -m[0]*m[5]*m[14]+m[0]*m[6]*m[13]+m[4]*m[1]*m[14]-m[4]*m[2]*m[13]-m[12]*m[1]*m[6]+m[12]*m[2]*m[5];
    inv[3]  = -m[1]*m[6]*m[11]+m[1]*m[7]*m[10]+m[5]*m[2]*m[11]-m[5]*m[3]*m[10]-m[9]*m[2]*m[7]+m[9]*m[3]*m[6];
    inv[7]  =  m[0]*m[6]*m[11]-m[0]*m[7]*m[10]-m[4]*m[2]*m[11]+m[4]*m[3]*m[10]+m[8]*m[2]*m[7]-m[8]*m[3]*m[6];
    inv[11] = -m[0]*m[5]*m[11]+m[0]*m[7]*m[9]+m[4]*m[1]*m[11]-m[4]*m[3]*m[9]-m[8]*m[1]*m[7]+m[8]*m[3]*m[5];
    inv[15] =  m[0]*m[5]*m[10]-m[0]*m[6]*m[9]-m[4]*m[1]*m[10]+m[4]*m[2]*m[9]+m[8]*m[1]*m[6]-m[8]*m[2]*m[5];
    float det = m[0]*inv[0]+m[1]*inv[4]+m[2]*inv[8]+m[3]*inv[12];
    det = 1.0f/det;
    for (int i = 0; i < 16; ++i) o[i] = inv[i]*det;
}

__device__ __forceinline__ void matmul4(const float* A, const float* Bm, float* Cm) {
    for (int i = 0; i < 4; ++i)
        for (int j = 0; j < 4; ++j) {
            float s = 0.f;
            for (int k = 0; k < 4; ++k) s += A[i*4+k]*Bm[k*4+j];
            Cm[i*4+j] = s;
        }
}

// mats layout in d_ws (floats): [b*9]  invK 3x3 ; [B_*9 + (f*B_+b)*12]  P rows 0..2
__global__ void prep_mats(const float* __restrict__ kK, const float* __restrict__ K,
                          const float* __restrict__ kpose, const float* __restrict__ poses,
                          float* __restrict__ mats) {
    int t = threadIdx.x;
    if (t < B_) {
        float inv[16];
        inv4x4(kK + t*16, inv);
        for (int r = 0; r < 3; ++r)
            for (int c = 0; c < 3; ++c)
                mats[t*9 + r*3 + c] = inv[r*4 + c];
    } else if (t < B_ + F_*B_) {
        int idx = t - B_;
        int f = idx / B_, b = idx % B_;
        float ip[16], T[16], PK[16];
        inv4x4(poses + (f*B_+b)*16, ip);
        matmul4(ip, kpose + b*16, T);
        matmul4(K + (f*B_+b)*16, T, PK);
        for (int i = 0; i < 12; ++i)
            mats[B_*9 + (f*B_+b)*12 + i] = PK[i];
    }
}

// -------------------------------------------------------------- warp helpers
__device__ __forceinline__ void proj_coords(float cx, float cy, float cz,
                                            float& sx, float& sy) {
    float zz = cz + 1e-7f;
    float px = cx / zz;
    float py = cy / zz;
    float gxn = px / (float)(W_-1);
    float gyn = py / (float)(H_-1);
    float grx = fminf(fmaxf((gxn - 0.5f)*2.0f, -2.0f), 2.0f);
    float gry = fminf(fmaxf((gyn - 0.5f)*2.0f, -2.0f), 2.0f);
    sx = (grx + 1.0f) * ((float)W_*0.5f) - 0.5f;
    sy = (gry + 1.0f) * ((float)H_*0.5f) - 0.5f;
}

// ---------------------------------------------------------------- main kernel
__global__ __launch_bounds__(256) void cv_main(
        const float* __restrict__ keyframe, const float* __restrict__ frames,
        const float* __restrict__ mats, float* __restrict__ sfcv,
        float* __restrict__ cv_acc, float* __restrict__ ws_acc) {
    __shared__ float sM[HALO*HALO*3];   // depth-independent ray vectors P33*K^-1*(x,y,1)
    __shared__ float sY[HALO*HALO*3];   // keyframe + 0.5 (reflect-padded)
    __shared__ float sSY[DWN*DWN*3];    // 3x3 sums of y
    __shared__ float sSY2[DWN*DWN*3];   // 3x3 sums of y*y
    __shared__ float sX[HALO*HALO*3];   // warped frame + 0.5 (per depth)
    __shared__ float sDW[DWN*DWN];      // channel-weighted SSIM diff
    __shared__ float sH[HALO*TIL];      // horizontal 3-sums (rows 0..17, pad rows 18,19 = 0)
    __shared__ float sSad[D_*256];      // per-depth sad per thread

    const int tid = threadIdx.x;
    const int z  = blockIdx.z;          // z = f*B_ + b
    const int b  = z % B_;
    const int ty0 = blockIdx.y * TIL;
    const int tx0 = blockIdx.x * TIL;
    const size_t HW = (size_t)H_ * W_;

    float ik[9], P[12];
    {
        const float* ikp = mats + b*9;
        const float* pp  = mats + B_*9 + z*12;
        #pragma unroll
        for (int i = 0; i < 9; ++i) ik[i] = ikp[i];
        #pragma unroll
        for (int i = 0; i < 12; ++i) P[i] = pp[i];
    }
    const float* kf = keyframe + (size_t)b * C_ * HW;
    const float* fr = frames   + (size_t)z * C_ * HW;

    // Phase 0: ray vectors + keyframe (reflect-padded)
    for (int pos = tid; pos < HALO*HALO; pos += 256) {
        int hy = pos / HALO, hx = pos % HALO;
        int gy = ty0 + hy - 2, gx = tx0 + hx - 2;
        int ry = gy < 0 ? -gy : (gy >= H_ ? 2*H_-2-gy : gy);
        int rx = gx < 0 ? -gx : (gx >= W_ ? 2*W_-2-gx : gx);
        float fx = (float)rx, fy = (float)ry;
        float cx = ik[0]*fx + ik[1]*fy + ik[2];
        float cy = ik[3]*fx + ik[4]*fy + ik[5];
        float cz = ik[6]*fx + ik[7]*fy + ik[8];
        sM[pos*3+0] = P[0]*cx + P[1]*cy + P[2]*cz;
        sM[pos*3+1] = P[4]*cx + P[5]*cy + P[6]*cz;
        sM[pos*3+2] = P[8]*cx + P[9]*cy + P[10]*cz;
        size_t off = (size_t)ry*W_ + rx;
        sY[pos*3+0] = kf[off]        + 0.5f;
        sY[pos*3+1] = kf[HW   + off] + 0.5f;
        sY[pos*3+2] = kf[2*HW + off] + 0.5f;
    }
    if (tid < 32) sH[DWN*TIL + tid] = 0.0f;   // zero pad rows 18,19 of sH
    __syncthreads();

    // Phase 0b: keyframe 3x3 window sums
    for (int pos = tid; pos < DWN*DWN; pos += 256) {
        int dy = pos / DWN, dx = pos % DWN;
        #pragma unroll
        for (int c = 0; c < 3; ++c) {
            float s = 0.f, s2 = 0.f;
            #pragma unroll
            for (int i = 0; i < 3; ++i)
                #pragma unroll
                for (int j = 0; j < 3; ++j) {
                    float v = sY[((dy+i)*HALO + (dx+j))*3 + c];
                    s += v; s2 += v*v;
                }
            sSY [pos*3+c] = s;
            sSY2[pos*3+c] = s2;
        }
    }
    __syncthreads();

    // thread -> pixel map matched to WMMA C-layout: lane l of wave w holds
    // element (row = w + 8*(l>=16), col = l&15) in acc component [w].
    const int lane = tid & 31;
    const int wv   = tid >> 5;
    const int ppx  = lane & 15;
    const int ppy  = wv + ((lane >> 4) << 3);
    const int gpy = ty0 + ppy, gpx = tx0 + ppx;
    const bool borderOK = (gpy >= 2 && gpy < H_-2 && gpx >= 2 && gpx < W_-2);
    const float om0 = sM[((ppy+2)*HALO + (ppx+2))*3+0];
    const float om1 = sM[((ppy+2)*HALO + (ppx+2))*3+1];
    const float om2 = sM[((ppy+2)*HALO + (ppx+2))*3+2];

    bool wmOK = true;
    float minsad = 1e30f;
    const float CW0 = 5.0f/32.0f, CW1 = 16.0f/32.0f, CW2 = 11.0f/32.0f;

    #pragma unroll 1
    for (int d = 0; d < D_; ++d) {
        float invd  = (1.0f/3.0f) + ((1.0f/80.0f) - (1.0f/3.0f)) * ((float)d * (1.0f/31.0f));
        float depth = 1.0f / invd;

        // ---- warp frame into sX (bilinear, zeros pad) ----
        for (int pos = tid; pos < HALO*HALO; pos += 256) {
            float cxp = depth*sM[pos*3+0] + P[3];
            float cyp = depth*sM[pos*3+1] + P[7];
            float czp = depth*sM[pos*3+2] + P[11];
            float sx, sy; proj_coords(cxp, cyp, czp, sx, sy);
            float x0f = floorf(sx), y0f = floorf(sy);
            int x0 = (int)x0f, y0 = (int)y0f, x1 = x0+1, y1 = y0+1;
            float wa = (x0f+1.0f-sx)*(y0f+1.0f-sy);
            float wb = (x0f+1.0f-sx)*(sy-y0f);
            float wc = (sx-x0f)*(y0f+1.0f-sy);
            float wd = (sx-x0f)*(sy-y0f);
            bool vx0 = (x0 >= 0) && (x0 < W_), vx1 = (x1 >= 0) && (x1 < W_);
            bool vy0 = (y0 >= 0) && (y0 < H_), vy1 = (y1 >= 0) && (y1 < H_);
            int xc0 = min(max(x0,0),W_-1), xc1 = min(max(x1,0),W_-1);
            int yc0 = min(max(y0,0),H_-1), yc1 = min(max(y1,0),H_-1);
            size_t ia = (size_t)yc0*W_+xc0, ib2 = (size_t)yc1*W_+xc0;
            size_t ic2 = (size_t)yc0*W_+xc1, id2 = (size_t)yc1*W_+xc1;
            #pragma unroll
            for (int c = 0; c < 3; ++c) {
                const float* pl = fr + (size_t)c*HW;
                float ga = (vx0 && vy0) ? pl[ia]  : 0.0f;
                float gb = (vx0 && vy1) ? pl[ib2] : 0.0f;
                float gc = (vx1 && vy0) ? pl[ic2] : 0.0f;
                float gd = (vx1 && vy1) ? pl[id2] : 0.0f;
                sX[pos*3+c] = ga*wa + gb*wb + gc*wc + gd*wd + 0.5f;
            }
        }
        __syncthreads();

        // ---- SSIM diff, channel-weighted, zero outside image ----
        for (int pos = tid; pos < DWN*DWN; pos += 256) {
            int dy = pos / DWN, dx = pos % DWN;
            int gy = ty0 + dy - 1, gx = tx0 + dx - 1;
            float dw = 0.0f;
            if (gy >= 0 && gy < H_ && gx >= 0 && gx < W_) {
                #pragma unroll
                for (int c = 0; c < 3; ++c) {
                    float sxv = 0.f, sxx = 0.f, sxy = 0.f;
                    #pragma unroll
                    for (int i = 0; i < 3; ++i)
                        #pragma unroll
                        for (int j = 0; j < 3; ++j) {
                            int q = ((dy+i)*HALO + (dx+j))*3 + c;
                            float xv = sX[q], yv = sY[q];
                            sxv += xv; sxx += xv*xv; sxy += xv*yv;
                        }
                    float mux = sxv * (1.0f/9.0f);
                    float muy = sSY [pos*3+c] * (1.0f/9.0f);
                    float sgx = sxx * (1.0f/9.0f) - mux*mux;
                    float sgy = sSY2[pos*3+c] * (1.0f/9.0f) - muy*muy;
                    float sgxy = sxy * (1.0f/9.0f) - mux*muy;
                    float nn = (2.0f*mux*muy + 1e-4f) * (2.0f*sgxy + 9e-4f);
                    float dd = (mux*mux + muy*muy + 1e-4f) * (sgx + sgy + 9e-4f);
                    float df = fminf(fmaxf((1.0f - nn/dd) * 0.5f, 0.0f), 1.0f);
                    dw += (c == 0 ? CW0 : (c == 1 ? CW1 : CW2)) * df;
                }
            }
            sDW[pos] = dw;
        }

        // ---- own-pixel warped border mask (analytic bilinear of indicator) ----
        {
            float cxp = depth*om0 + P[3];
            float cyp = depth*om1 + P[7];
            float czp = depth*om2 + P[11];
            float sx, sy; proj_coords(cxp, cyp, czp, sx, sy);
            float x0f = floorf(sx), y0f = floorf(sy);
            int x0 = (int)x0f, y0 = (int)y0f, x1 = x0+1, y1 = y0+1;
            float wa = (x0f+1.0f-sx)*(y0f+1.0f-sy);
            float wb = (x0f+1.0f-sx)*(sy-y0f);
            float wc = (sx-x0f)*(y0f+1.0f-sy);
            float wd = (sx-x0f)*(sy-y0f);
            float ma = (x0 >= 2 && x0 <= W_-3 && y0 >= 2 && y0 <= H_-3) ? 1.f : 0.f;
            float mb = (x0 >= 2 && x0 <= W_-3 && y1 >= 2 && y1 <= H_-3) ? 1.f : 0.f;
            float mc = (x1 >= 2 && x1 <= W_-3 && y0 >= 2 && y0 <= H_-3) ? 1.f : 0.f;
            float md = (x1 >= 2 && x1 <= W_-3 && y1 >= 2 && y1 <= H_-3) ? 1.f : 0.f;
            float wm = ma*wa + mb*wb + mc*wc + md*wd;
            wmOK = wmOK && (wm != 0.0f);
        }
        __syncthreads();

        // ---- horizontal 3-sum ----
        for (int pos = tid; pos < DWN*TIL; pos += 256) {
            int r = pos / TIL, cc = pos % TIL;
            sH[r*TIL+cc] = sDW[r*DWN+cc] + sDW[r*DWN+cc+1] + sDW[r*DWN+cc+2];
        }
        __syncthreads();

        // ---- vertical 3-sum as banded matmul: sad = Rv(16x20) x H(20x16), via
        //      chained V_WMMA_F32_16X16X4_F32 (each wave computes the full tile) ----
        v8f acc = {0.f,0.f,0.f,0.f,0.f,0.f,0.f,0.f};
        const int khalf = (lane >> 4) << 1;   // lanes 16..31 hold K+2 (A & B layouts)
        const int am = lane & 15;             // A-matrix row M = lane&15
        #pragma unroll
        for (int k0 = 0; k0 < HALO; k0 += 4) {
            int kA = k0 + khalf;
            int kB = kA + 1;
            v2f av, bv;
            av.x = (kA >= am && kA <= am+2) ? 1.0f : 0.0f;  // Rv[m,k]=1 for k in [m,m+2]
            av.y = (kB >= am && kB <= am+2) ? 1.0f : 0.0f;
            bv.x = sH[kA*TIL + ppx];
            bv.y = sH[kB*TIL + ppx];
            acc = __builtin_amdgcn_wmma_f32_16x16x4_f32(
                false, av, false, bv, (short)0, acc, false, false);
        }
        float sad = acc[wv] * (1.0f/9.0f);
        sSad[d*256 + tid] = sad;
        minsad = fminf(minsad, sad);
        __syncthreads();
    }

    // ---- per-pixel depth reduction + outputs ----
    float wmask = (borderOK && wmOK) ? 1.0f : 0.0f;
    float sum = 0.0f;
    #pragma unroll 1
    for (int d = 0; d < D_; ++d) {
        float e = sSad[d*256 + tid] - minsad;
        sum += expf(-10.0f * e * e);
    }
    float weight = (1.0f - (sum - 1.0f) * (1.0f/31.0f)) * wmask;
    size_t pix = (size_t)gpy * W_ + gpx;
    #pragma unroll 1
    for (int d = 0; d < D_; ++d) {
        float s = sSad[d*256 + tid];
        sfcv[((size_t)z * D_ + d) * HW + pix] = (1.0f - 2.0f*s) * wmask;
        atomicAdd(&cv_acc[((size_t)b * D_ + d) * HW + pix], s * weight);
    }
    atomicAdd(&ws_acc[(size_t)b * HW + pix], weight);
}

// --------------------------------------------------------------- finalize cv
__global__ void finalize_cv(float* __restrict__ cv, const float* __restrict__ ws_acc) {
    size_t i = (size_t)blockIdx.x * 256 + threadIdx.x;
    const size_t total = (size_t)B_ * D_ * H_ * W_;
    if (i >= total) return;
    const size_t HW = (size_t)H_ * W_;
    size_t b = i / ((size_t)D_ * HW);
    size_t pix = i % HW;
    float wsv = ws_acc[b * HW + pix];
    float v = cv[i];
    cv[i] = (wsv == 0.0f) ? 0.0f : 1.0f - 2.0f * v / wsv;
}

extern "C" void kernel_launch(void* const* d_in, const int* in_sizes, int n_in,
                              void* d_out, int out_size, void* d_ws, size_t ws_size,
                              hipStream_t stream) {
    (void)in_sizes; (void)n_in; (void)out_size; (void)ws_size;
    const float* keyframe = (const float*)d_in[0];
    const float* frames   = (const float*)d_in[1];
    const float* kK       = (const float*)d_in[2];
    const float* K        = (const float*)d_in[3];
    const float* kpose    = (const float*)d_in[4];
    const float* poses    = (const float*)d_in[5];

    float* out   = (float*)d_out;
    float* cv    = out;                                       // [B,D,H,W]
    float* sfcv  = out + (size_t)B_ * D_ * H_ * W_;           // [F,B,D,H,W]
    float* mats  = (float*)d_ws;                              // 66 floats
    float* wsacc = (float*)((char*)d_ws + 4096);              // [B,H,W]

    hipMemsetAsync(cv, 0, (size_t)B_ * D_ * H_ * W_ * sizeof(float), stream);
    hipMemsetAsync(wsacc, 0, (size_t)B_ * H_ * W_ * sizeof(float), stream);

    prep_mats<<<1, 32, 0, stream>>>(kK, K, kpose, poses, mats);

    dim3 grid(W_/TIL, H_/TIL, B_*F_);
    cv_main<<<grid, 256, 0, stream>>>(keyframe, frames, mats, sfcv, cv, wsacc);

    size_t total = (size_t)B_ * D_ * H_ * W_;
    finalize_cv<<<(unsigned)((total + 255) / 256), 256, 0, stream>>>(cv, wsacc);
}